// D_constraint1_17600775979893
// MI455X (gfx1250) — compile-verified
//
#include <hip/hip_runtime.h>
#include <math.h>

typedef __attribute__((ext_vector_type(2))) float v2f;
typedef __attribute__((ext_vector_type(8))) float v8f;

#define NCOLS    256
#define NBLOCKS  1024
#define T1       256   // kernel 1: 8 waves
#define T2       512   // kernel 2: 16 waves (one 16-column group per wave)

// ---------------------------------------------------------------------------
// Pass 1: per-block partial column sums of squares.
// Block b handles rows [b*rpb, b*rpb+rpb). Thread t: row-subgroup g = t>>6
// (uniform per wave32), column-quad cq = t&63 (float4 of columns 4cq..4cq+3).
// A wave's 32 lanes load 32 consecutive float4s = 512 contiguous bytes
// (global_load_b128, perfectly coalesced). LDS combines the 4 subgroups and
// one 256-float partial row per block is written to d_ws.
// ---------------------------------------------------------------------------
__global__ __launch_bounds__(T1) void colsq_partial_kernel(
    const float* __restrict__ d, float* __restrict__ partial,
    int nrows, int rows_per_block)
{
    const int t  = threadIdx.x;
    const int g  = t >> 6;      // 0..3, uniform within a wave
    const int cq = t & 63;      // which float4 within the 256-float row
    const int row0 = blockIdx.x * rows_per_block + g;

    const float4* dp = reinterpret_cast<const float4*>(d);
    float4 acc = make_float4(0.f, 0.f, 0.f, 0.f);

    #pragma unroll 4
    for (int k = 0; k < rows_per_block; k += 4) {
        const int row = row0 + k;
        if (row < nrows) {                       // uniform per wave
            float4 v = dp[row * (NCOLS / 4) + cq];
            acc.x = fmaf(v.x, v.x, acc.x);
            acc.y = fmaf(v.y, v.y, acc.y);
            acc.z = fmaf(v.z, v.z, acc.z);
            acc.w = fmaf(v.w, v.w, acc.w);
        }
    }

    __shared__ float lds[4 * NCOLS];
    reinterpret_cast<float4*>(lds)[g * (NCOLS / 4) + cq] = acc;
    __syncthreads();

    if (t < NCOLS) {
        float s = lds[t] + lds[NCOLS + t] + lds[2 * NCOLS + t] + lds[3 * NCOLS + t];
        partial[blockIdx.x * NCOLS + t] = s;
    }
}

// ---------------------------------------------------------------------------
// Pass 2: WMMA reduction of the P x 256 partial matrix.
// Wave w owns columns [16w, 16w+16). A = ones(16x4) so
//   C[m][n] += sum over 4 partial rows of B[k][n]  (all m rows identical).
// Lane L supplies column N = L%16; the two B VGPRs x two half-waves carry the
// four partial rows k..k+3 (any K-slot bijection is correct since A is ones).
// After the chain, c[0] in every lane = col_sq of its column.
// ---------------------------------------------------------------------------
__global__ __launch_bounds__(T2) void finalize_kernel(
    const float* __restrict__ partial, float* __restrict__ out, int P)
{
    const int t    = threadIdx.x;
    const int wave = t >> 5;           // 0..15 -> column group
    const int lane = t & 31;
    const int col  = wave * 16 + (lane & 15);
    const int krow = lane >> 4;        // 0 for lanes 0-15, 1 for lanes 16-31

    v2f a; a[0] = 1.0f; a[1] = 1.0f;   // A = ones, layout-independent
    v8f c = {0.f, 0.f, 0.f, 0.f, 0.f, 0.f, 0.f, 0.f};

    #pragma unroll 4
    for (int k = 0; k < P; k += 4) {
        v2f b;
        b[0] = partial[(k + krow)     * NCOLS + col];
        b[1] = partial[(k + 2 + krow) * NCOLS + col];
        // D = A(16x4 ones) * B(4x16 partial rows) + C  -> running column sums
        c = __builtin_amdgcn_wmma_f32_16x16x4_f32(
                /*neg_a=*/false, a, /*neg_b=*/false, b,
                /*c_mod=*/(short)0, c, /*reuse_a=*/false, /*reuse_b=*/false);
    }

    float s  = c[0];                                   // col_sq[col] (all lanes)
    float v  = (lane < 16) ? (s - 1.0f) * (s - 1.0f)   // count each column once
                           : 0.0f;

    // wave32 shuffle reduction
    #pragma unroll
    for (int off = 16; off > 0; off >>= 1)
        v += __shfl_xor(v, off, 32);

    __shared__ float red[T2 / 32];
    if (lane == 0) red[wave] = v;
    __syncthreads();

    if (t == 0) {
        float tot = 0.f;
        #pragma unroll
        for (int i = 0; i < T2 / 32; ++i) tot += red[i];
        out[0] = 0.001f * sqrtf(tot);
    }
}

// ---------------------------------------------------------------------------
extern "C" void kernel_launch(void* const* d_in, const int* in_sizes, int n_in,
                              void* d_out, int out_size, void* d_ws, size_t ws_size,
                              hipStream_t stream)
{
    const float* d = (const float*)d_in[0];
    const int total = in_sizes[0];          // 262144 * 256
    const int nrows = total / NCOLS;        // 262144

    float* partial = (float*)d_ws;          // NBLOCKS * 256 floats = 1 MiB

    int rpb = (nrows + NBLOCKS - 1) / NBLOCKS;
    rpb = (rpb + 3) & ~3;                   // multiple of 4 for row subgroups

    colsq_partial_kernel<<<NBLOCKS, T1, 0, stream>>>(d, partial, nrows, rpb);
    finalize_kernel<<<1, T2, 0, stream>>>(partial, (float*)d_out, NBLOCKS);
}